// OurModel_40948218200804
// MI455X (gfx1250) — compile-verified
//
#include <hip/hip_runtime.h>
#include <hip/hip_bf16.h>
#include <math.h>

// Problem constants (match reference setup_inputs)
#define KSZ   10
#define NKER  32
#define QD    32
#define NHEADS 2
#define DH    16
#define NHID  512
#define EMBD  10
#define KNNB  10
#define Bz    32
#define Tz    5000
#define Nz    512
#define Sz    500        // T / KS
#define QT    32         // ceil(S/16)
#define SPAD  512        // padded S for 16x16 tiling

typedef float v2f __attribute__((ext_vector_type(2)));
typedef float v8f __attribute__((ext_vector_type(8)));

// CDNA5 fp32 WMMA: D(16x16) = A(16x4) * B(4x16) + C   -> v_wmma_f32_16x16x4_f32
static __device__ __forceinline__ v8f wmma4(v2f a, v2f b, v8f c) {
  return __builtin_amdgcn_wmma_f32_16x16x4_f32(false, a, false, b, (short)0, c,
                                               false, false);
}

// ---------------------------------------------------------------- conv (stride KS)
__global__ void k_conv(const float* __restrict__ x, const float* __restrict__ cw,
                       const float* __restrict__ cb, float* __restrict__ val) {
  int i = blockIdx.x * blockDim.x + threadIdx.x;       // ((b*S+s)*32 + k)
  if (i >= Bz * Sz * NKER) return;
  int k = i & 31;
  int bs = i >> 5;
  int s = bs % Sz;
  int b = bs / Sz;
  const float* xp = x + (size_t)b * Tz + s * KSZ;
  float acc = cb[k];
#pragma unroll
  for (int j = 0; j < KSZ; j++) acc += xp[j] * cw[k * KSZ + j];
  val[i] = acc;
}

// ------------------------------------------------- q_pos / Q,K,V projections (1 wave per (b,s))
__global__ void __launch_bounds__(32)
k_proj(const float* __restrict__ val, const float* __restrict__ mqw,
       const float* __restrict__ mqb, const float* __restrict__ wq,
       const float* __restrict__ bq, const float* __restrict__ wk,
       const float* __restrict__ bk, const float* __restrict__ wv,
       const float* __restrict__ bv, float* __restrict__ Qm,
       float* __restrict__ Km, float* __restrict__ Vm) {
  int bs = blockIdx.x;
  int s = bs % Sz;
  int c = threadIdx.x;
  __shared__ float vl[32], vr[32], vb[32], qp[32];
  vl[c] = (s > 0) ? val[(size_t)(bs - 1) * 32 + c] : 0.f;
  vr[c] = (s < Sz - 1) ? val[(size_t)(bs + 1) * 32 + c] : 0.f;
  vb[c] = val[(size_t)bs * 32 + c];
  __syncthreads();
  float q = mqb[c];
#pragma unroll 8
  for (int i = 0; i < 32; i++)
    q += vl[i] * mqw[i * QD + c] + vr[i] * mqw[(32 + i) * QD + c];
  // positional encoding: pe[s,2i]=sin(s*div_i), pe[s,2i+1]=cos(s*div_i)
  int half = c >> 1;
  float div = expf(-(float)(2 * half) * 0.28782313662425572f); // ln(1e4)/32
  float ang = (float)s * div;
  float pe = (c & 1) ? cosf(ang) : sinf(ang);
  qp[c] = fmaxf(q, 0.f) + pe;
  __syncthreads();
  float aq = bq[c], ak = bk[c], av = bv[c];
#pragma unroll 8
  for (int i = 0; i < 32; i++) {
    float p = qp[i];
    aq += p * wq[i * QD + c];
    ak += p * wk[i * QD + c];
    av += fmaxf(vb[i], 0.f) * wv[i * QD + c];
  }
  Qm[(size_t)bs * 32 + c] = aq;
  Km[(size_t)bs * 32 + c] = ak;
  Vm[(size_t)bs * 32 + c] = av;
}

// ------------------------------------------- pooled: mean of 10 contiguous floats of ctx3
// flat identity: pooled[o] = mean(ctx3[o*10 .. o*10+9]); grid covers B*S*S exactly.
// Data movement done by the Tensor Data Mover (one 10240-byte 1-row tile per block)
// into LDS, then a per-thread 10-tap reduce.
__global__ void __launch_bounds__(256)
k_pool(const float* __restrict__ ctx, float* __restrict__ pooled) {
  __shared__ float buf[2560];
  int base = blockIdx.x * 256;
  const float* src = ctx + (size_t)base * KSZ;
  int t = threadIdx.x;
#if __has_builtin(__builtin_amdgcn_tensor_load_to_lds) && \
    __has_builtin(__builtin_amdgcn_s_wait_tensorcnt)
  typedef unsigned int u32x4 __attribute__((ext_vector_type(4)));
  typedef int i32x4 __attribute__((ext_vector_type(4)));
  typedef int i32x8 __attribute__((ext_vector_type(8)));
  // TDM is EXEC-independent: restrict issue to wave 0 via a scalar branch.
  int wid = __builtin_amdgcn_readfirstlane((int)(t >> 5));
  if (wid == 0) {
    unsigned long long ga = (unsigned long long)(uintptr_t)src;
    unsigned int ldsa = (unsigned int)(uintptr_t)(
        (__attribute__((address_space(3))) float*)buf);
    u32x4 g0;
    g0.x = 1u;                                         // count=1 (valid user D#)
    g0.y = ldsa;                                       // lds_addr (bytes)
    g0.z = (unsigned int)ga;                           // global_addr[31:0]
    g0.w = (unsigned int)((ga >> 32) & 0x01FFFFFFu)    // global_addr[56:32]
           | (2u << 30);                               // type=2 ("image")
    i32x8 g1;
    g1[0] = 0x00020000;              // workgroup_mask=0, data_size=2 (4 bytes)
    g1[1] = (int)(2560u << 16);      // tensor_dim0[15:0]=2560
    g1[2] = (int)(1u << 16);         // tensor_dim0[31:16]=0, tensor_dim1[15:0]=1
    g1[3] = (int)(2560u << 16);      // tensor_dim1[31:16]=0, tile_dim0=2560
    g1[4] = 1;                       // tile_dim1=1, tile_dim2=0
    g1[5] = 2560;                    // tensor_dim0_stride[31:0]
    g1[6] = 0;                       // stride0[47:32]=0, stride1[15:0]=0
    g1[7] = 0;                       // stride1[47:16]=0
    i32x4 z4 = {};
    i32x8 z8 = {};
    // 6-arg variant (clang-23 / therock headers): groups 2,3 + extra group, cpol
    __builtin_amdgcn_tensor_load_to_lds(g0, g1, z4, z4, z8, 0);
    __builtin_amdgcn_s_wait_tensorcnt(0);
  }
  __syncthreads();
#else
  __builtin_prefetch(src + 2560, 0, 1);
#pragma unroll
  for (int i = 0; i < 10; i++) buf[t + i * 256] = src[t + i * 256];
  __syncthreads();
#endif
  float acc = 0.f;
#pragma unroll
  for (int j = 0; j < 10; j++) acc += buf[t * 10 + j];
  pooled[base + t] = acc * 0.1f;
}

// ------------------------------------------- KNN neighbor selection per batch element
__global__ void __launch_bounds__(64)
k_nbr(const float* __restrict__ emb, const int* __restrict__ idx,
      int* __restrict__ nidx, float* __restrict__ nw) {
  int b = blockIdx.x;
  int t = threadIdx.x;
  __shared__ float wvv[Nz];
  __shared__ float es[EMBD];
  __shared__ float rv[64];
  __shared__ int ri[64];
  int self = idx[b];
  if (t < EMBD) es[t] = emb[(size_t)self * EMBD + t];
  __syncthreads();
  for (int n = t; n < Nz; n += 64) {
    float ss = 0.f;
#pragma unroll
    for (int e = 0; e < EMBD; e++) {
      float d = es[e] - emb[(size_t)n * EMBD + e];
      ss += d * d;
    }
    float d = sqrtf(fmaxf(ss, 1e-24f)) + 0.001f;
    wvv[n] = (n == self) ? -1.f : expf(-d);   // exclude self (it is the max, w=1)
  }
  __syncthreads();
  for (int k = 0; k < KNNB; k++) {
    float bw = -2.f;
    int bi = 0;
    for (int n = t; n < Nz; n += 64) {
      float w = wvv[n];
      if (w > bw) { bw = w; bi = n; }
    }
    rv[t] = bw; ri[t] = bi;
    __syncthreads();
    for (int off = 32; off > 0; off >>= 1) {
      if (t < off && rv[t + off] > rv[t]) { rv[t] = rv[t + off]; ri[t] = ri[t + off]; }
      __syncthreads();
    }
    if (t == 0) {
      nidx[b * KNNB + k] = ri[0];
      nw[b * KNNB + k] = rv[0];
      wvv[ri[0]] = -1.f;
    }
    __syncthreads();
  }
}

// --------------------------- fold deconv_w with out_w: dw[i][j] = sum_o dcw[i][o][j]*ow[o]
__global__ void __launch_bounds__(64)
k_dw(const float* __restrict__ dcw, const float* __restrict__ dcb,
     const float* __restrict__ ow, const float* __restrict__ ob,
     float* __restrict__ dw) {
  int t = threadIdx.x;
  for (int e = t; e < NKER * KSZ; e += 64) {
    int i = e / KSZ, j = e % KSZ;
    float a = 0.f;
#pragma unroll 8
    for (int o = 0; o < NKER; o++) a += dcw[(i * NKER + o) * KSZ + j] * ow[o];
    dw[e] = a;
  }
  if (t == 0) {
    float a = ob[0];
#pragma unroll 8
    for (int o = 0; o < NKER; o++) a += dcb[o] * ow[o];
    dw[NKER * KSZ] = a;
  }
}

// ---------------------------------------------------------------- attention core
// one wave per (qtile, head, batch): WMMA scores -> LDS, softmax, WMMA P@V
// USEP selects the precomputed-pooled path vs on-the-fly ctx3 averaging at compile time.
template <int USEP>
__global__ void __launch_bounds__(32)
k_attn(const float* __restrict__ Qm, const float* __restrict__ Km,
       const float* __restrict__ Vm, const float* __restrict__ pooled,
       const float* __restrict__ ctx, float* __restrict__ oat) {
  int qt = blockIdx.x, h = blockIdx.y, b = blockIdx.z;
  int l = threadIdx.x, lo = l & 15, hi = l >> 4;
  int s0 = qt * 16;
  __shared__ float sc[16][SPAD];

  // A fragments for Q tile (M=16 x K=dh=16), 4 k-steps of 4
  v2f aq[4];
#pragma unroll
  for (int k4 = 0; k4 < 4; k4++) {
    int row = s0 + lo;
    int col = h * DH + k4 * 4 + 2 * hi;
    if (row < Sz) {
      const float* p = Qm + ((size_t)(b * Sz + row)) * QD + col;
      aq[k4].x = p[0]; aq[k4].y = p[1];
    } else { aq[k4].x = 0.f; aq[k4].y = 0.f; }
  }

  // scores = Q K^T / 4 + pooled + diag(-1e9), tiled over 32 key tiles
  for (int kt = 0; kt < QT; kt++) {
    v8f c = {};
#pragma unroll
    for (int k4 = 0; k4 < 4; k4++) {
      v2f bf;
      int col = kt * 16 + lo;
      int d = h * DH + k4 * 4 + 2 * hi;
      if (col < Sz) {
        const float* p = Km + ((size_t)(b * Sz + col)) * QD + d;
        bf.x = p[0]; bf.y = p[1];
      } else { bf.x = 0.f; bf.y = 0.f; }
      c = wmma4(aq[k4], bf, c);
    }
#pragma unroll
    for (int r = 0; r < 8; r++) {
      int m = r + 8 * hi;
      int qrow = s0 + m, kcol = kt * 16 + lo;
      float bias;
      if (qrow >= Sz || kcol >= Sz) bias = -1e9f;
      else {
        float p;
        if (USEP) {
          p = pooled[((size_t)(b * Sz + qrow)) * Sz + kcol];
        } else {
          const float* cp = ctx + ((size_t)(b * Sz + qrow)) * Tz + kcol * KSZ;
          float a0 = 0.f;
#pragma unroll
          for (int j = 0; j < KSZ; j++) a0 += cp[j];
          p = a0 * 0.1f;
        }
        bias = p + ((qrow == kcol) ? -1e9f : 0.f);
      }
      sc[m][kcol] = c[r] * 0.25f + bias;
    }
  }
  __syncthreads();

  // row softmax over 512 (padded cols are -1e9 -> exactly 0 after normalization)
  for (int m = 0; m < 16; m++) {
    float vals[16];
    float mx = -1e30f;
#pragma unroll
    for (int i = 0; i < 16; i++) {
      float v = sc[m][l + i * 32];
      vals[i] = v;
      mx = fmaxf(mx, v);
    }
#pragma unroll
    for (int off = 16; off > 0; off >>= 1) mx = fmaxf(mx, __shfl_xor(mx, off, 32));
    float sum = 0.f;
#pragma unroll
    for (int i = 0; i < 16; i++) {
      float e = __expf(vals[i] - mx);
      vals[i] = e;
      sum += e;
    }
#pragma unroll
    for (int off = 16; off > 0; off >>= 1) sum += __shfl_xor(sum, off, 32);
    float inv = 1.f / sum;
#pragma unroll
    for (int i = 0; i < 16; i++) sc[m][l + i * 32] = vals[i] * inv;
  }
  __syncthreads();

  // o = P @ V  (16x512 * 512x16), K loop of 128 WMMA steps
  v8f co = {};
  for (int kk = 0; kk < SPAD / 4; kk++) {
    int kb = kk * 4 + 2 * hi;
    v2f a;
    a.x = sc[lo][kb];
    a.y = sc[lo][kb + 1];
    v2f bf;
    bf.x = (kb < Sz) ? Vm[((size_t)(b * Sz + kb)) * QD + h * DH + lo] : 0.f;
    bf.y = (kb + 1 < Sz) ? Vm[((size_t)(b * Sz + kb + 1)) * QD + h * DH + lo] : 0.f;
    co = wmma4(a, bf, co);
  }
#pragma unroll
  for (int r = 0; r < 8; r++) {
    int row = s0 + r + 8 * hi;
    if (row < Sz) oat[((size_t)(b * Sz + row)) * QD + h * DH + lo] = co[r];
  }
}

// -------------------------------- output projection + FFN + hidden (2 waves per q-tile)
__global__ void __launch_bounds__(64)
k_post(const float* __restrict__ oat, const float* __restrict__ wo,
       const float* __restrict__ bo, const float* __restrict__ w1,
       const float* __restrict__ b1, const float* __restrict__ w2,
       const float* __restrict__ b2, float* __restrict__ hid) {
  int bq = blockIdx.x;
  int qt = bq % QT, b = bq / QT;
  int s0 = qt * 16;
  int t = threadIdx.x, wv_ = t >> 5, l = t & 31, lo = l & 15, hi = l >> 4;
  __shared__ float ot[16][32];
  __shared__ float o2[16][32];
  __shared__ float hb[16][NHID];

  for (int e = t; e < 16 * 32; e += 64) {
    int m = e >> 5, c = e & 31;
    int row = s0 + m;
    ot[m][c] = (row < Sz) ? oat[((size_t)(b * Sz + row)) * QD + c] : 0.f;
  }
  __syncthreads();

  // o2 = ot @ wo + bo : wave wv_ computes its 16-col tile
  {
    v8f c = {};
#pragma unroll
    for (int k8 = 0; k8 < 8; k8++) {
      int kb = k8 * 4 + 2 * hi;
      v2f a; a.x = ot[lo][kb]; a.y = ot[lo][kb + 1];
      v2f bf;
      int n = wv_ * 16 + lo;
      bf.x = wo[kb * QD + n];
      bf.y = wo[(kb + 1) * QD + n];
      c = wmma4(a, bf, c);
    }
    int n = wv_ * 16 + lo;
    float bb = bo[n];
#pragma unroll
    for (int r = 0; r < 8; r++) o2[r + 8 * hi][n] = c[r] + bb;
  }
  __syncthreads();

  // hb = relu(o2 @ w1 + b1) : wave wv_ computes 256 cols (16 tiles)
  for (int nt = 0; nt < 16; nt++) {
    int n0 = wv_ * 256 + nt * 16;
    v8f c = {};
#pragma unroll
    for (int k8 = 0; k8 < 8; k8++) {
      int kb = k8 * 4 + 2 * hi;
      v2f a; a.x = o2[lo][kb]; a.y = o2[lo][kb + 1];
      v2f bf;
      int n = n0 + lo;
      bf.x = w1[kb * NHID + n];
      bf.y = w1[(kb + 1) * NHID + n];
      c = wmma4(a, bf, c);
    }
    int n = n0 + lo;
    float bb = b1[n];
#pragma unroll
    for (int r = 0; r < 8; r++) hb[r + 8 * hi][n] = fmaxf(c[r] + bb, 0.f);
  }
  __syncthreads();

  // hidden = relu(o2 + hb @ w2 + b2) : wave wv_ computes its 16-col tile (K=512)
  {
    v8f c = {};
    for (int kk = 0; kk < NHID / 4; kk++) {
      int kb = kk * 4 + 2 * hi;
      v2f a; a.x = hb[lo][kb]; a.y = hb[lo][kb + 1];
      v2f bf;
      int n = wv_ * 16 + lo;
      bf.x = w2[kb * QD + n];
      bf.y = w2[(kb + 1) * QD + n];
      c = wmma4(a, bf, c);
    }
    int n = wv_ * 16 + lo;
    float bb = b2[n];
#pragma unroll
    for (int r = 0; r < 8; r++) {
      int m = r + 8 * hi;
      int row = s0 + m;
      if (row < Sz) {
        float v = o2[m][n] + c[r] + bb;
        hid[((size_t)(b * Sz + row)) * QD + n] = fmaxf(v, 0.f);
      }
    }
  }
}

// ----------------------- fused siblings + folded deconv + masked abs-error reduction
__global__ void __launch_bounds__(256)
k_final(const float* __restrict__ x, const float* __restrict__ res,
        const int* __restrict__ mask, const int* __restrict__ idx,
        const float* __restrict__ stdv, const float* __restrict__ hid,
        const float* __restrict__ dw, const int* __restrict__ nidx,
        const float* __restrict__ nw, const float* __restrict__ ow,
        float* __restrict__ acc) {
  int g = blockIdx.x * 256 + threadIdx.x;    // b*T + t, grid covers exactly B*T
  int b = g / Tz;
  int tt = g % Tz;
  int s = tt / KSZ, j = tt % KSZ;
  float mean = dw[NKER * KSZ];               // folded deconv_b.out_w + out_b
  const float* hrow = hid + ((size_t)(b * Sz + s)) * QD;
#pragma unroll 8
  for (int i = 0; i < NKER; i++) mean += hrow[i] * dw[i * KSZ + j];
  // siblings
  const float* rrow = res + ((size_t)(b * Tz + tt)) * Nz;
  float wsum = 0.f, wss = 0.f, sum = 0.f, sum2 = 0.f;
#pragma unroll
  for (int k = 0; k < KNNB; k++) {
    float sv = rrow[nidx[b * KNNB + k]];
    float wk_ = nw[b * KNNB + k];
    float tw = (sv == 0.f) ? wk_ * 1e-9f : wk_;
    wsum += tw;
    wss += tw * sv;
    sum += sv;
    sum2 += sv * sv;
  }
  float wm = wss / wsum;
  float mu = sum * (1.f / KNNB);
  float var = (sum2 - (float)KNNB * mu * mu) * (1.f / (KNNB - 1));
  float sd = sqrtf(fmaxf(var, 0.f));
  mean += wsum * ow[NKER + 0] + wm * ow[NKER + 1] + sd * ow[NKER + 2];
  float temp = fabsf(x[g] - mean) * stdv[idx[b]];
  float sel = (mask[g] > 0) ? 1.f : 0.f;

  __shared__ float r1[256], r2[256];
  int th = threadIdx.x;
  r1[th] = temp * sel;
  r2[th] = sel;
  __syncthreads();
  for (int off = 128; off > 0; off >>= 1) {
    if (th < off) { r1[th] += r1[th + off]; r2[th] += r2[th + off]; }
    __syncthreads();
  }
  if (th == 0) {
    atomicAdd(&acc[0], r1[0]);
    atomicAdd(&acc[1], r2[0]);
  }
}

__global__ void k_zero(float* __restrict__ acc) {
  if (threadIdx.x < 2) acc[threadIdx.x] = 0.f;
}
__global__ void k_div(const float* __restrict__ acc, float* __restrict__ out) {
  out[0] = acc[0] / fmaxf(acc[1], 1.f);
}

// ================================================================ host launch
extern "C" void kernel_launch(void* const* d_in, const int* in_sizes, int n_in,
                              void* d_out, int out_size, void* d_ws, size_t ws_size,
                              hipStream_t stream) {
  const float* in_series = (const float*)d_in[0];
  const float* residuals = (const float*)d_in[1];
  const float* ctx3      = (const float*)d_in[2];
  const int*   mask      = (const int*)d_in[3];
  const int*   idx       = (const int*)d_in[4];
  const float* embed_w   = (const float*)d_in[5];
  const float* stdv      = (const float*)d_in[6];
  const float* conv_w    = (const float*)d_in[7];
  const float* conv_b    = (const float*)d_in[8];
  const float* deconv_w  = (const float*)d_in[9];
  const float* deconv_b  = (const float*)d_in[10];
  const float* mq_w      = (const float*)d_in[11];
  const float* mq_b      = (const float*)d_in[12];
  const float* wq        = (const float*)d_in[13];
  const float* bq        = (const float*)d_in[14];
  const float* wk        = (const float*)d_in[15];
  const float* bk        = (const float*)d_in[16];
  const float* wv        = (const float*)d_in[17];
  const float* bv        = (const float*)d_in[18];
  const float* wo        = (const float*)d_in[19];
  const float* bo        = (const float*)d_in[20];
  const float* ffn_w1    = (const float*)d_in[21];
  const float* ffn_b1    = (const float*)d_in[22];
  const float* ffn_w2    = (const float*)d_in[23];
  const float* ffn_b2    = (const float*)d_in[24];
  const float* out_w     = (const float*)d_in[25];
  const float* out_b     = (const float*)d_in[26];

  char* ws = (char*)d_ws;
  size_t off = 0;
  auto carve = [&](size_t bytes) -> char* {
    char* p = ws + off;
    off = (off + bytes + 255) & ~(size_t)255;
    return p;
  };
  const size_t rowsB = (size_t)Bz * Sz * NKER * sizeof(float);  // 2 MB
  float* val  = (float*)carve(rowsB);
  float* Qm   = (float*)carve(rowsB);
  float* Km   = (float*)carve(rowsB);
  float* Vm   = (float*)carve(rowsB);
  float* oat  = (float*)carve(rowsB);
  float* hid  = (float*)carve(rowsB);
  float* dwv  = (float*)carve(336 * sizeof(float));
  int*   nidx = (int*)  carve(Bz * KNNB * sizeof(int));
  float* nw   = (float*)carve(Bz * KNNB * sizeof(float));
  float* acc  = (float*)carve(2 * sizeof(float));
  // pooled (32 MB) only if workspace is large enough; else compute on-the-fly
  size_t pooledBytes = (size_t)Bz * Sz * Sz * sizeof(float);
  int use_pooled = (ws_size >= off + pooledBytes) ? 1 : 0;
  float* pooled = (float*)(ws + off);

  k_zero<<<1, 32, 0, stream>>>(acc);
  k_conv<<<(Bz * Sz * NKER + 255) / 256, 256, 0, stream>>>(in_series, conv_w,
                                                           conv_b, val);
  k_proj<<<Bz * Sz, 32, 0, stream>>>(val, mq_w, mq_b, wq, bq, wk, bk, wv, bv,
                                     Qm, Km, Vm);
  k_nbr<<<Bz, 64, 0, stream>>>(embed_w, idx, nidx, nw);
  k_dw<<<1, 64, 0, stream>>>(deconv_w, deconv_b, out_w, out_b, dwv);
  if (use_pooled) {
    k_pool<<<(Bz * Sz * Sz) / 256, 256, 0, stream>>>(ctx3, pooled);
    k_attn<1><<<dim3(QT, NHEADS, Bz), 32, 0, stream>>>(Qm, Km, Vm, pooled, ctx3,
                                                       oat);
  } else {
    k_attn<0><<<dim3(QT, NHEADS, Bz), 32, 0, stream>>>(Qm, Km, Vm, pooled, ctx3,
                                                       oat);
  }
  k_post<<<Bz * QT, 64, 0, stream>>>(oat, wo, bo, ffn_w1, ffn_b1, ffn_w2,
                                     ffn_b2, hid);
  k_final<<<(Bz * Tz) / 256, 256, 0, stream>>>(in_series, residuals, mask, idx,
                                               stdv, hid, dwv, nidx, nw, out_w,
                                               acc);
  k_div<<<1, 1, 0, stream>>>(acc, (float*)d_out);
}